// GraphLearner_68513318306086
// MI455X (gfx1250) — compile-verified
//
#include <hip/hip_runtime.h>
#include <hip/hip_bf16.h>
#include <math.h>

typedef __attribute__((ext_vector_type(2))) float v2f;
typedef __attribute__((ext_vector_type(8))) float v8f;

#define DIM 64
#define COS_EPS 1e-6f

// ---------------------------------------------------------------------------
// Phase 1: per-node inverse norm. 16 lanes cooperate on one 64-float row
// (float4 per lane, coalesced 256B row read), reduce via xor-shuffles.
// 100K rows * 256B = 25.6MB streamed once; result (400KB) parks in L2.
// ---------------------------------------------------------------------------
__global__ __launch_bounds__(256) void node_invnorm_kernel(
    const float* __restrict__ embs, float* __restrict__ invn, int n_nodes) {
  int t = blockIdx.x * blockDim.x + threadIdx.x;
  int node = t >> 4;
  int q = t & 15;
  if (node >= n_nodes) return;
  const float4 v = *(const float4*)(embs + (size_t)node * DIM + q * 4);
  float ss = v.x * v.x + v.y * v.y + v.z * v.z + v.w * v.w;
  // reduce across the 16-lane group (masks < 16 stay inside the group)
  ss += __shfl_xor(ss, 1);
  ss += __shfl_xor(ss, 2);
  ss += __shfl_xor(ss, 4);
  ss += __shfl_xor(ss, 8);
  if (q == 0) invn[node] = 1.0f / fmaxf(sqrtf(ss), COS_EPS);
}

// Defensive scalar read: harness may store python scalar 1 as i32/i64/f32.
__device__ inline float read_temperature(const void* p) {
  int iv = *(const int*)p;                 // low 4 bytes
  float fv = __int_as_float(iv);
  float af = fabsf(fv);
  bool plausible_float = (af > 1e-6f) && (af < 1e6f);
  return plausible_float ? fv : (float)iv; // i64(1)/i32(1) -> 1.0f
}

// ---------------------------------------------------------------------------
// Phase 2: one wave32 = one tile of 16 edges.
//   A (16x64, f32) = f1 rows of the 16 edges
//   B (64x16, f32) = f2 rows transposed
//   G = A x B via 16 chained V_WMMA_F32_16X16X4_F32; diag(G) = per-edge dots.
// f32 A(16x4) / B(4x16) VGPR layout (ISA 7.12.2): lanes 0-15 hold K={4j,4j+1},
// lanes 16-31 hold K={4j+2,4j+3}  ->  per-step operand is one float2 load at
// row_base + 4j + 2*(lane>=16), identical formula for A and B.
// Diagonal (m,m): lane m, C[m] for m<8; lane m+16, C[m-8] for m>=8.
// ---------------------------------------------------------------------------
__global__ __launch_bounds__(256) void edge_cosine_wmma_kernel(
    const float* __restrict__ embs, const float* __restrict__ invn,
    const int* __restrict__ src, const int* __restrict__ dst,
    const void* __restrict__ temp_ptr, float* __restrict__ out,
    long long n_edges) {
  const int lane = threadIdx.x & 31;
  const int wave_in_block = threadIdx.x >> 5;
  const long long tile = (long long)blockIdx.x * (blockDim.x >> 5) + wave_in_block;
  const long long base = tile * 16;
  if (base >= n_edges) return;   // whole wave exits together (EXEC stays full)

  const int m = lane & 15;       // edge-in-tile this lane loads for
  const int koff = (lane >> 4) * 2;  // 0 for lanes 0-15, 2 for lanes 16-31
  long long e = base + m;
  const bool edge_valid = (e < n_edges);
  const long long esafe = edge_valid ? e : 0;

  const int r1 = src[esafe];
  const int r2 = dst[esafe];
  const float* p1 = embs + (size_t)r1 * DIM + koff;  // A operand source
  const float* p2 = embs + (size_t)r2 * DIM + koff;  // B operand source

  v2f a[16], b[16];
#pragma unroll
  for (int j = 0; j < 16; ++j) {
    a[j] = *(const v2f*)(p1 + 4 * j);
    b[j] = *(const v2f*)(p2 + 4 * j);
  }

  v8f c = {};
#pragma unroll
  for (int j = 0; j < 16; ++j) {
    // D = A(16x4) * B(4x16) + C, exact fp32
    c = __builtin_amdgcn_wmma_f32_16x16x4_f32(
        /*neg_a=*/false, a[j], /*neg_b=*/false, b[j],
        /*c_mod=*/(short)0, c, /*reuse_a=*/false, /*reuse_b=*/false);
  }

  // Diagonal extraction: lanes 0-7 own edges 0-7 (C[lane]),
  // lanes 24-31 own edges 8-15 (C[lane-24]). Both match this lane's m.
  const bool active = (lane < 8) || (lane >= 24);
  const int idx = lane & 7;
  float dotv = c[0];
#pragma unroll
  for (int i = 1; i < 8; ++i) dotv = (idx == i) ? c[i] : dotv;

  if (active && edge_valid) {
    const float inv_t = 1.0f / read_temperature(temp_ptr);
    const float cosv = dotv * invn[r1] * invn[r2];
    const float x = cosv * inv_t;
    const float s = 1.0f / (1.0f + expf(-x));        // sigmoid
    out[e] = (s <= 0.5f) ? 0.0f : s;                  // straight-through hard
  }
}

extern "C" void kernel_launch(void* const* d_in, const int* in_sizes, int n_in,
                              void* d_out, int out_size, void* d_ws, size_t ws_size,
                              hipStream_t stream) {
  const float* embs = (const float*)d_in[0];
  const int* edge_index = (const int*)d_in[1];
  const void* temp = d_in[2];
  float* out = (float*)d_out;

  const int n_nodes = in_sizes[0] / DIM;
  const long long n_edges = (long long)in_sizes[1] / 2;
  const int* src = edge_index;
  const int* dst = edge_index + n_edges;

  float* invn = (float*)d_ws;  // n_nodes floats of scratch

  // Phase 1: per-node inverse norms (16 threads per node)
  {
    long long threads = (long long)n_nodes * 16;
    int blocks = (int)((threads + 255) / 256);
    node_invnorm_kernel<<<blocks, 256, 0, stream>>>(embs, invn, n_nodes);
  }

  // Phase 2: WMMA edge tiles (8 waves/block -> 8 tiles of 16 edges per block)
  {
    long long tiles = (n_edges + 15) / 16;
    long long blocks = (tiles + 7) / 8;
    edge_cosine_wmma_kernel<<<(int)blocks, 256, 0, stream>>>(
        embs, invn, src, dst, temp, out, n_edges);
  }
}